// GraphSAGE_49520972923235
// MI455X (gfx1250) — compile-verified
//
#include <hip/hip_runtime.h>

#define N_NODES 100000
#define N_EDGES 800000
#define DIM     128
#define NTILES_M (N_NODES / 16)   // 6250
#define NTILES_N (DIM / 16)       // 8

typedef __attribute__((ext_vector_type(2))) float v2f;
typedef __attribute__((ext_vector_type(8))) float v8f;

// ---------------------------------------------------------------------------
// Zero a float region (grid-stride).
// ---------------------------------------------------------------------------
__global__ void zero_f32(float* __restrict__ p, int n) {
    int i = blockIdx.x * blockDim.x + threadIdx.x;
    int stride = gridDim.x * blockDim.x;
    for (; i < n; i += stride) p[i] = 0.0f;
}

// ---------------------------------------------------------------------------
// Degree count: cnt[dst[e]] += 1   (computed once; graph reused both layers)
// ---------------------------------------------------------------------------
__global__ void degree_kernel(const int* __restrict__ dst, float* __restrict__ cnt) {
    int e = blockIdx.x * blockDim.x + threadIdx.x;
    if (e < N_EDGES) atomicAdd(&cnt[dst[e]], 1.0f);
}

// ---------------------------------------------------------------------------
// Edge scatter: agg[dst] += h[src].  One wave (32 lanes) per edge, each lane
// moves 4 contiguous floats (float4 load -> 4 global_atomic_add_f32).
// All of h/agg is L2-resident (192MB L2 >> 51MB per array).
// ---------------------------------------------------------------------------
__global__ void scatter_kernel(const float* __restrict__ h,
                               const int* __restrict__ src,
                               const int* __restrict__ dst,
                               float* __restrict__ agg) {
    int gid  = blockIdx.x * blockDim.x + threadIdx.x;
    int e    = gid >> 5;
    int lane = gid & 31;
    if (e >= N_EDGES) return;
    int s = src[e];
    int d = dst[e];
    const float4 v = *((const float4*)(h + (size_t)s * DIM) + lane);
    float* out = agg + (size_t)d * DIM + lane * 4;
    atomicAdd(out + 0, v.x);
    atomicAdd(out + 1, v.y);
    atomicAdd(out + 2, v.z);
    atomicAdd(out + 3, v.w);
}

// ---------------------------------------------------------------------------
// Fused SAGEConv layer:
//   out = act( (agg/max(cnt,1)) @ Wl^T + b + h @ Wr^T )
// One wave per 16x16 output tile; K=128 swept in steps of 4 with chained
// v_wmma_f32_16x16x4_f32 (full f32 precision; compute is nowhere near the
// bottleneck on MI455X, so no reason to drop to bf16).
//
// Fragment layouts per CDNA5 ISA 7.12.2:
//   A (16x4):  lane holds row m=lane&15, k = (lane>>4)*2 + {0,1}
//   B (4x16):  lane holds col n=lane&15, k = (lane>>4)*2 + {0,1}
//   C/D(16x16):vgpr v: row = (lane>>4)*8 + v, col = lane&15
// ---------------------------------------------------------------------------
__global__ void sage_gemm_relu(const float* __restrict__ agg,
                               const float* __restrict__ cnt,
                               const float* __restrict__ h,
                               const float* __restrict__ Wl,
                               const float* __restrict__ Wr,
                               const float* __restrict__ bias,
                               float* __restrict__ out,
                               int do_relu) {
    int gid    = blockIdx.x * blockDim.x + threadIdx.x;
    int wave   = gid >> 5;
    int lane   = gid & 31;
    if (wave >= NTILES_M * NTILES_N) return;

    int row_tile = wave / NTILES_N;        // consecutive waves in a block share A rows
    int col_tile = wave % NTILES_N;
    int node0 = row_tile * 16;
    int j0    = col_tile * 16;

    int m    = lane & 15;
    int half = lane >> 4;
    int koff = half * 2;

    const float* arow = agg + (size_t)(node0 + m) * DIM;  // A source (mean)
    const float* hrow = h   + (size_t)(node0 + m) * DIM;  // A source (self)
    const float* wlc  = Wl  + (size_t)(j0 + m) * DIM;     // B source: B[k][n]=W[n][k]
    const float* wrc  = Wr  + (size_t)(j0 + m) * DIM;

    float invc = 1.0f / fmaxf(cnt[node0 + m], 1.0f);      // mean scale for this lane's A row

    float bval = bias[j0 + (lane & 15)];                  // bias depends only on column
    v8f acc;
    #pragma unroll
    for (int v = 0; v < 8; ++v) acc[v] = bval;

    #pragma unroll 4
    for (int kk = 0; kk < DIM; kk += 4) {
        v2f a1 = *(const v2f*)(arow + kk + koff);
        a1 = a1 * invc;                                   // form mean on the fly
        v2f b1 = *(const v2f*)(wlc + kk + koff);
        acc = __builtin_amdgcn_wmma_f32_16x16x4_f32(
            /*neg_a=*/false, a1, /*neg_b=*/false, b1,
            /*c_mod=*/(short)0, acc, /*reuse_a=*/false, /*reuse_b=*/false);

        v2f a2 = *(const v2f*)(hrow + kk + koff);
        v2f b2 = *(const v2f*)(wrc + kk + koff);
        acc = __builtin_amdgcn_wmma_f32_16x16x4_f32(
            false, a2, false, b2, (short)0, acc, false, false);
    }

    int col = j0 + (lane & 15);
    #pragma unroll
    for (int v = 0; v < 8; ++v) {
        int row = node0 + half * 8 + v;
        float val = acc[v];
        if (do_relu) val = fmaxf(val, 0.0f);
        out[(size_t)row * DIM + col] = val;
    }
}

// ---------------------------------------------------------------------------
// Final FC: out[n] = h2[n,:] . Wfc + bfc   (one thread per node, float4 loads)
// ---------------------------------------------------------------------------
__global__ void fc_kernel(const float* __restrict__ h,
                          const float* __restrict__ Wfc,
                          const float* __restrict__ bfc,
                          float* __restrict__ out) {
    int n = blockIdx.x * blockDim.x + threadIdx.x;
    if (n >= N_NODES) return;
    const float4* hp = (const float4*)(h + (size_t)n * DIM);
    const float4* wp = (const float4*)Wfc;
    float s = 0.0f;
    #pragma unroll
    for (int i = 0; i < DIM / 4; ++i) {
        float4 a = hp[i];
        float4 b = wp[i];
        s += a.x * b.x + a.y * b.y + a.z * b.z + a.w * b.w;
    }
    out[n] = s + bfc[0];
}

// ---------------------------------------------------------------------------
// Launch sequence
// ---------------------------------------------------------------------------
extern "C" void kernel_launch(void* const* d_in, const int* in_sizes, int n_in,
                              void* d_out, int out_size, void* d_ws, size_t ws_size,
                              hipStream_t stream) {
    const float* x    = (const float*)d_in[0];
    const int*   src  = (const int*)d_in[1];              // edge_index[0]
    const int*   dst  = ((const int*)d_in[1]) + N_EDGES;  // edge_index[1]
    const float* W1l  = (const float*)d_in[2];
    const float* b1   = (const float*)d_in[3];
    const float* W1r  = (const float*)d_in[4];
    const float* W2l  = (const float*)d_in[5];
    const float* b2   = (const float*)d_in[6];
    const float* W2r  = (const float*)d_in[7];
    const float* Wfc  = (const float*)d_in[8];
    const float* bfc  = (const float*)d_in[9];
    float* out = (float*)d_out;

    // Workspace layout (floats): cnt | agg | h1 | h2  (~154 MB total)
    float* ws  = (float*)d_ws;
    float* cnt = ws;
    float* agg = cnt + N_NODES;
    float* h1  = agg + (size_t)N_NODES * DIM;
    float* h2  = h1  + (size_t)N_NODES * DIM;

    const int feat_elems = N_NODES * DIM;

    dim3 blk(256);
    dim3 grid_zero_all((N_NODES + feat_elems + 255) / 256 > 4096 ? 4096
                       : (N_NODES + feat_elems + 255) / 256);
    dim3 grid_zero_agg(4096);
    dim3 grid_deg((N_EDGES + 255) / 256);
    dim3 grid_scat((N_EDGES * 32 + 255) / 256);           // 32 lanes per edge
    dim3 grid_gemm((NTILES_M * NTILES_N * 32 + 255) / 256); // 1 wave per 16x16 tile
    dim3 grid_fc((N_NODES + 255) / 256);

    // ---- shared degree counts ----
    zero_f32<<<grid_zero_all, blk, 0, stream>>>(cnt, N_NODES + feat_elems); // cnt+agg
    degree_kernel<<<grid_deg, blk, 0, stream>>>(dst, cnt);

    // ---- layer 1 ----
    scatter_kernel<<<grid_scat, blk, 0, stream>>>(x, src, dst, agg);
    sage_gemm_relu<<<grid_gemm, blk, 0, stream>>>(agg, cnt, x, W1l, W1r, b1, h1, 1);

    // ---- layer 2 ----
    zero_f32<<<grid_zero_agg, blk, 0, stream>>>(agg, feat_elems);
    scatter_kernel<<<grid_scat, blk, 0, stream>>>(h1, src, dst, agg);
    sage_gemm_relu<<<grid_gemm, blk, 0, stream>>>(agg, cnt, h1, W2l, W2r, b2, h2, 1);

    // ---- final FC ----
    fc_kernel<<<grid_fc, blk, 0, stream>>>(h2, Wfc, bfc, out);
}